// DROPGNN_1623497638676
// MI455X (gfx1250) — compile-verified
//
#include <hip/hip_runtime.h>

typedef __attribute__((ext_vector_type(2))) float v2f;
typedef __attribute__((ext_vector_type(8))) float v8f;

static inline int ceildiv(int a, int b) { return (a + b - 1) / b; }

// ---------------- degree / normalization ----------------

__global__ void k_deg_init(float* __restrict__ deg, int n) {
  int i = blockIdx.x * blockDim.x + threadIdx.x;
  if (i < n) deg[i] = 1.0f;  // self-loop contribution
}

__global__ void k_deg_count(const int* __restrict__ dst, float* __restrict__ deg, int e) {
  int i = blockIdx.x * blockDim.x + threadIdx.x;
  if (i < e) atomicAdd(&deg[dst[i]], 1.0f);
}

__global__ void k_rsqrt_inplace(float* __restrict__ d, int n) {
  int i = blockIdx.x * blockDim.x + threadIdx.x;
  if (i < n) {
    float v = d[i];
    d[i] = (v > 0.0f) ? rsqrtf(v) : 0.0f;
  }
}

// ---------------- WMMA fp32 GEMM: M = A(nrows x K) @ W(K x ncols) ----------------
// One wave computes a 16x16 tile of M using V_WMMA_F32_16X16X4_F32 chained over K.
// blockDim.x = 32 * (ncols/16); wave w owns column tile w; blockIdx.x owns row tile.
// Requires nrows % 16 == 0, K % 4 == 0, ncols % 16 == 0 (true here: 50000,128,{128,64}).

__global__ void k_gemm_wmma_f32(const float* __restrict__ A, const float* __restrict__ W,
                                float* __restrict__ M, int K, int ncols) {
  const int wave = threadIdx.x >> 5;
  const int lane = threadIdx.x & 31;
  const int lrow = lane & 15;
  const int koff = (lane >= 16) ? 2 : 0;   // lane halves hold K=0/1 vs K=2/3
  const int row0 = blockIdx.x * 16;
  const int col0 = wave * 16;

  const float* __restrict__ arow = A + (size_t)(row0 + lrow) * K + koff;
  const float* __restrict__ bcol = W + (size_t)koff * ncols + (col0 + lrow);

  v8f acc = {};
  for (int k = 0; k < K; k += 4) {
    v2f a, b;
    a.x = arow[k];
    a.y = arow[k + 1];
    b.x = bcol[(size_t)k * ncols];
    b.y = bcol[(size_t)(k + 1) * ncols];
    acc = __builtin_amdgcn_wmma_f32_16x16x4_f32(
        /*neg_a=*/false, a, /*neg_b=*/false, b,
        /*c_mod=*/(short)0, acc, /*reuse_a=*/false, /*reuse_b=*/false);
  }

  // C/D layout: lanes 0-15 -> rows M=0..7 in acc[0..7]; lanes 16-31 -> rows M=8..15.
  const int rbase = row0 + ((lane >= 16) ? 8 : 0);
  float* __restrict__ out = M + (size_t)col0 + lrow;
#pragma unroll
  for (int r = 0; r < 8; ++r) {
    out[(size_t)(rbase + r) * ncols] = acc[r];
  }
}

// ---------------- aggregation ----------------

// agg[i][c] = M[i][c] * dinv[i]^2   (self-loop term; also fully initializes agg)
__global__ void k_agg_init_self(const float* __restrict__ M, const float* __restrict__ dinv,
                                float* __restrict__ agg, int n, int ncols) {
  long long i = (long long)blockIdx.x * blockDim.x + threadIdx.x;
  long long total = (long long)n * ncols;
  if (i >= total) return;
  int node = (int)(i >> (ncols == 128 ? 7 : 6));
  float di = dinv[node];
  agg[i] = M[i] * (di * di);
}

// One wave per edge: agg[dst] += M[src] * dinv[src]*dinv[dst]
template <int PER>  // floats per lane = ncols/32
__global__ void k_edge_scatter(const float* __restrict__ M, const int* __restrict__ src,
                               const int* __restrict__ dst, const float* __restrict__ dinv,
                               float* __restrict__ agg, int nedges, int ncols) {
  int wid = (blockIdx.x * blockDim.x + threadIdx.x) >> 5;
  int lane = threadIdx.x & 31;
  if (wid >= nedges) return;
  int s = src[wid];
  int d = dst[wid];
  float nrm = dinv[s] * dinv[d];
  const float* __restrict__ mp = M + (size_t)s * ncols + lane * PER;
  float* __restrict__ ap = agg + (size_t)d * ncols + lane * PER;
#pragma unroll
  for (int j = 0; j < PER; ++j) {
    atomicAdd(&ap[j], mp[j] * nrm);
  }
}

// h = relu(agg + b), in place
__global__ void k_bias_relu(float* __restrict__ agg, const float* __restrict__ b,
                            int n, int ncols) {
  long long i = (long long)blockIdx.x * blockDim.x + threadIdx.x;
  long long total = (long long)n * ncols;
  if (i >= total) return;
  float v = agg[i] + b[(int)(i & (ncols - 1))];
  agg[i] = (v > 0.0f) ? v : 0.0f;
}

// log_softmax over 64 classes, adding bias first; one wave per node, 2 elems/lane
__global__ void k_logsoftmax64(float* __restrict__ logits, const float* __restrict__ bias,
                               int n) {
  int wid = (blockIdx.x * blockDim.x + threadIdx.x) >> 5;
  int lane = threadIdx.x & 31;
  if (wid >= n) return;
  float* __restrict__ row = logits + (size_t)wid * 64;
  float v0 = row[lane] + bias[lane];
  float v1 = row[lane + 32] + bias[lane + 32];
  float m = fmaxf(v0, v1);
#pragma unroll
  for (int o = 16; o > 0; o >>= 1) m = fmaxf(m, __shfl_xor(m, o, 32));
  float s = __expf(v0 - m) + __expf(v1 - m);
#pragma unroll
  for (int o = 16; o > 0; o >>= 1) s += __shfl_xor(s, o, 32);
  float ls = __logf(s);
  row[lane]      = v0 - m - ls;
  row[lane + 32] = v1 - m - ls;
}

// ---------------- driver ----------------

extern "C" void kernel_launch(void* const* d_in, const int* in_sizes, int n_in,
                              void* d_out, int out_size, void* d_ws, size_t ws_size,
                              hipStream_t stream) {
  const float* x  = (const float*)d_in[0];
  const int*   ei = (const int*)d_in[1];
  const float* W1 = (const float*)d_in[2];
  const float* b1 = (const float*)d_in[3];
  const float* W2 = (const float*)d_in[4];
  const float* b2 = (const float*)d_in[5];
  const float* Wf = (const float*)d_in[6];
  const float* bf = (const float*)d_in[7];
  float* out = (float*)d_out;

  const int HID  = in_sizes[3];            // 128
  const int FIN  = in_sizes[2] / HID;      // 128
  const int N    = in_sizes[0] / FIN;      // 50000
  const int E    = in_sizes[1] / 2;        // 625000
  const int C    = in_sizes[7];            // 64
  const int* src = ei;
  const int* dst = ei + E;

  // workspace layout (floats): dinv[N] | M[N*HID] | bufA[N*HID] | bufB[N*HID]
  float* dinv = (float*)d_ws;
  float* M    = dinv + N;
  float* bufA = M + (size_t)N * HID;
  float* bufB = bufA + (size_t)N * HID;

  const int T = 256;

  // ---- normalization coefficients ----
  k_deg_init<<<ceildiv(N, T), T, 0, stream>>>(dinv, N);
  k_deg_count<<<ceildiv(E, T), T, 0, stream>>>(dst, dinv, E);
  k_rsqrt_inplace<<<ceildiv(N, T), T, 0, stream>>>(dinv, N);

  const int rowTiles = N / 16;                 // 3125
  const long long NH = (long long)N * HID;
  const long long NC = (long long)N * C;

  // ---- layer 1: h1 = relu(agg(x @ W1) + b1) -> bufA ----
  k_gemm_wmma_f32<<<rowTiles, 32 * (HID / 16), 0, stream>>>(x, W1, M, FIN, HID);
  k_agg_init_self<<<(int)((NH + T - 1) / T), T, 0, stream>>>(M, dinv, bufA, N, HID);
  k_edge_scatter<4><<<ceildiv(E, T / 32), T, 0, stream>>>(M, src, dst, dinv, bufA, E, HID);
  k_bias_relu<<<(int)((NH + T - 1) / T), T, 0, stream>>>(bufA, b1, N, HID);

  // ---- layer 2: h2 = relu(agg(h1 @ W2) + b2) -> bufB ----
  k_gemm_wmma_f32<<<rowTiles, 32 * (HID / 16), 0, stream>>>(bufA, W2, M, HID, HID);
  k_agg_init_self<<<(int)((NH + T - 1) / T), T, 0, stream>>>(M, dinv, bufB, N, HID);
  k_edge_scatter<4><<<ceildiv(E, T / 32), T, 0, stream>>>(M, src, dst, dinv, bufB, E, HID);
  k_bias_relu<<<(int)((NH + T - 1) / T), T, 0, stream>>>(bufB, b2, N, HID);

  // ---- layer 3: logits = agg(h2 @ Wf) + bf, then log_softmax -> d_out ----
  k_gemm_wmma_f32<<<rowTiles, 32 * (C / 16), 0, stream>>>(bufB, Wf, M, HID, C);
  k_agg_init_self<<<(int)((NC + T - 1) / T), T, 0, stream>>>(M, dinv, out, N, C);
  k_edge_scatter<2><<<ceildiv(E, T / 32), T, 0, stream>>>(M, src, dst, dinv, out, E, C);
  k_logsoftmax64<<<ceildiv(N, T / 32), T, 0, stream>>>(out, bf, N);
}